// GAT_35880156791351
// MI455X (gfx1250) — compile-verified
//
#include <hip/hip_runtime.h>
#include <hip/hip_bf16.h>

typedef __attribute__((ext_vector_type(2))) float v2f;
typedef __attribute__((ext_vector_type(8))) float v8f;

// ---------------------------------------------------------------- helpers

__device__ __forceinline__ float lrelu(float v) { return v > 0.0f ? v : 0.2f * v; }

__device__ __forceinline__ float atomicMaxF(float* addr, float val) {
    unsigned int* ua = (unsigned int*)addr;
    unsigned int old = *ua;
    while (true) {
        float cur = __uint_as_float(old);
        if (cur >= val) break;
        unsigned int assumed = old;
        old = atomicCAS(ua, assumed, __float_as_uint(val));
        if (old == assumed) break;
    }
    return __uint_as_float(old);
}

__global__ void fill_f32(float* __restrict__ p, float v, long long n) {
    long long i = blockIdx.x * (long long)blockDim.x + threadIdx.x;
    if (i < n) p[i] = v;
}

// ---------------------------------------------------------------- GEMM 1: H1 = X[N,128] @ W1[128,128]
// 8 waves / block, wave w owns column tile w (16 cols). One block per 16-row stripe.
__global__ void gemm1_wmma(const float* __restrict__ X, const float* __restrict__ W,
                           float* __restrict__ H, long long N) {
    const int lane    = threadIdx.x & 31;
    const int colTile = threadIdx.x >> 5;          // 0..7
    const int m       = lane & 15;                 // row within tile (A) / col within tile (B,C)
    const int kb      = (lane >> 4) << 1;          // K sub-offset: 0 for lanes 0-15, 2 for 16-31

    long long row  = (long long)blockIdx.x * 16 + m;
    long long lrow = row < N ? row : (N - 1);      // clamp loads; EXEC stays all-ones for WMMA
    const float* xrow = X + lrow * 128;
    const float* wcol = W + colTile * 16 + m;

    v8f acc = {};
#pragma unroll
    for (int i = 0; i < 32; ++i) {
        const int k = i * 4;
        v2f a, b;
        a.x = xrow[k + kb];
        a.y = xrow[k + kb + 1];
        b.x = wcol[(long long)(k + kb) * 128];
        b.y = wcol[(long long)(k + kb + 1) * 128];
        acc = __builtin_amdgcn_wmma_f32_16x16x4_f32(false, a, false, b,
                                                    (short)0, acc, false, false);
    }

    const int colBase = colTile * 16 + m;
    const long long rbase = (long long)blockIdx.x * 16 + ((lane >> 4) << 3);
#pragma unroll
    for (int j = 0; j < 8; ++j) {
        long long r = rbase + j;
        if (r < N) H[r * 128 + colBase] = acc[j];
    }
}

// ---------------------------------------------------------------- GEMM 2: H2 = A[N,128] @ W2[128,32]
// 2 waves / block (32 output cols), one block per 16-row stripe.
__global__ void gemm2_wmma(const float* __restrict__ X, const float* __restrict__ W,
                           float* __restrict__ H, long long N) {
    const int lane    = threadIdx.x & 31;
    const int colTile = threadIdx.x >> 5;          // 0..1
    const int m       = lane & 15;
    const int kb      = (lane >> 4) << 1;

    long long row  = (long long)blockIdx.x * 16 + m;
    long long lrow = row < N ? row : (N - 1);
    const float* xrow = X + lrow * 128;
    const float* wcol = W + colTile * 16 + m;

    v8f acc = {};
#pragma unroll
    for (int i = 0; i < 32; ++i) {
        const int k = i * 4;
        v2f a, b;
        a.x = xrow[k + kb];
        a.y = xrow[k + kb + 1];
        b.x = wcol[(long long)(k + kb) * 32];
        b.y = wcol[(long long)(k + kb + 1) * 32];
        acc = __builtin_amdgcn_wmma_f32_16x16x4_f32(false, a, false, b,
                                                    (short)0, acc, false, false);
    }

    const int colBase = colTile * 16 + m;
    const long long rbase = (long long)blockIdx.x * 16 + ((lane >> 4) << 3);
#pragma unroll
    for (int j = 0; j < 8; ++j) {
        long long r = rbase + j;
        if (r < N) H[r * 32 + colBase] = acc[j];
    }
}

// ---------------------------------------------------------------- attention logits (per node)
__global__ void logits1(const float* __restrict__ H, const float* __restrict__ a_src,
                        const float* __restrict__ a_dst, float* __restrict__ als,
                        float* __restrict__ ald, long long N) {
    long long n = blockIdx.x * (long long)blockDim.x + threadIdx.x;
    if (n >= N) return;
    const float* hrow = H + n * 128;
#pragma unroll
    for (int h = 0; h < 4; ++h) {
        float s = 0.0f, d = 0.0f;
#pragma unroll
        for (int c = 0; c < 32; ++c) {
            float v = hrow[h * 32 + c];
            s += v * a_src[h * 32 + c];
            d += v * a_dst[h * 32 + c];
        }
        als[n * 4 + h] = s;
        ald[n * 4 + h] = d;
    }
}

__global__ void logits2(const float* __restrict__ H, const float* __restrict__ a_src,
                        const float* __restrict__ a_dst, float* __restrict__ als,
                        float* __restrict__ ald, long long N) {
    long long n = blockIdx.x * (long long)blockDim.x + threadIdx.x;
    if (n >= N) return;
    const float* hrow = H + n * 32;
    float s = 0.0f, d = 0.0f;
#pragma unroll
    for (int c = 0; c < 32; ++c) {
        float v = hrow[c];
        s += v * a_src[c];
        d += v * a_dst[c];
    }
    als[n] = s;
    ald[n] = d;
}

// ---------------------------------------------------------------- layer-1 edge phase (4 heads)
// thread = (edge, head); edges e<E come from edge_index (int64), e>=E are self-loops
__global__ void edge_max1(const long long* __restrict__ ei, const float* __restrict__ als,
                          const float* __restrict__ ald, float* __restrict__ mbuf,
                          long long E, long long N) {
    long long idx = blockIdx.x * (long long)blockDim.x + threadIdx.x;
    if (idx >= (E + N) * 4) return;
    long long e = idx >> 2;
    int h = (int)(idx & 3);
    long long s, d;
    if (e < E) { s = ei[e]; d = ei[E + e]; } else { s = d = e - E; }
    float v = lrelu(als[s * 4 + h] + ald[d * 4 + h]);
    atomicMaxF(&mbuf[d * 4 + h], v);
}

// per (edge, head): unnormalized softmax weight -> pbuf, accumulate denominator
__global__ void edge_p1(const long long* __restrict__ ei, const float* __restrict__ als,
                        const float* __restrict__ ald, const float* __restrict__ mbuf,
                        float* __restrict__ pbuf, float* __restrict__ den,
                        long long E, long long N) {
    long long idx = blockIdx.x * (long long)blockDim.x + threadIdx.x;
    if (idx >= (E + N) * 4) return;
    long long e = idx >> 2;
    int h = (int)(idx & 3);
    long long s, d;
    if (e < E) { s = ei[e]; d = ei[E + e]; } else { s = d = e - E; }
    float v = lrelu(als[s * 4 + h] + ald[d * 4 + h]);
    float p = __expf(v - mbuf[d * 4 + h]);
    pbuf[idx] = p;
    atomicAdd(&den[d * 4 + h], p);
}

// per (edge, channel): coalesced gather of H[src] + coalesced atomic scatter to acc[dst]
// a wave covers 32 contiguous channels of ONE edge -> 1 coalesced load + 1 coalesced atomic
__global__ void edge_scatter1(const long long* __restrict__ ei, const float* __restrict__ pbuf,
                              const float* __restrict__ H, float* __restrict__ acc,
                              long long E, long long N) {
    long long idx = blockIdx.x * (long long)blockDim.x + threadIdx.x;
    if (idx >= (E + N) * 128) return;
    long long e = idx >> 7;
    int c = (int)(idx & 127);
    int h = c >> 5;
    long long s, d;
    if (e < E) { s = ei[e]; d = ei[E + e]; } else { s = d = e - E; }
    float p = pbuf[e * 4 + h];
    atomicAdd(&acc[d * 128 + c], p * H[s * 128 + c]);
}

// normalize by softmax denominator, add bias, ELU (in place -> layer-2 input)
__global__ void norm1(float* __restrict__ acc, const float* __restrict__ den,
                      const float* __restrict__ b1, long long N) {
    long long idx = blockIdx.x * (long long)blockDim.x + threadIdx.x;
    if (idx >= N * 128) return;
    long long n = idx >> 7;
    int c = (int)(idx & 127);
    int h = c >> 5;
    float dn = fmaxf(den[n * 4 + h], 1e-16f);
    float v = acc[idx] / dn + b1[c];
    acc[idx] = v > 0.0f ? v : (__expf(v) - 1.0f);   // ELU
}

// ---------------------------------------------------------------- layer-2 edge phase (1 head)
__global__ void edge_max2(const long long* __restrict__ ei, const float* __restrict__ als,
                          const float* __restrict__ ald, float* __restrict__ mbuf,
                          long long E, long long N) {
    long long idx = blockIdx.x * (long long)blockDim.x + threadIdx.x;
    if (idx >= E + N) return;
    long long s, d;
    if (idx < E) { s = ei[idx]; d = ei[E + idx]; } else { s = d = idx - E; }
    float v = lrelu(als[s] + ald[d]);
    atomicMaxF(&mbuf[d], v);
}

__global__ void edge_p2(const long long* __restrict__ ei, const float* __restrict__ als,
                        const float* __restrict__ ald, const float* __restrict__ mbuf,
                        float* __restrict__ pbuf, float* __restrict__ den,
                        long long E, long long N) {
    long long idx = blockIdx.x * (long long)blockDim.x + threadIdx.x;
    if (idx >= E + N) return;
    long long s, d;
    if (idx < E) { s = ei[idx]; d = ei[E + idx]; } else { s = d = idx - E; }
    float v = lrelu(als[s] + ald[d]);
    float p = __expf(v - mbuf[d]);
    pbuf[idx] = p;
    atomicAdd(&den[d], p);
}

__global__ void edge_scatter2(const long long* __restrict__ ei, const float* __restrict__ pbuf,
                              const float* __restrict__ H2, float* __restrict__ out,
                              long long E, long long N) {
    long long idx = blockIdx.x * (long long)blockDim.x + threadIdx.x;
    if (idx >= (E + N) * 32) return;
    long long e = idx >> 5;
    int c = (int)(idx & 31);
    long long s, d;
    if (e < E) { s = ei[e]; d = ei[E + e]; } else { s = d = e - E; }
    float p = pbuf[e];
    atomicAdd(&out[d * 32 + c], p * H2[s * 32 + c]);
}

__global__ void norm2(float* __restrict__ out, const float* __restrict__ den,
                      const float* __restrict__ b2, long long N) {
    long long idx = blockIdx.x * (long long)blockDim.x + threadIdx.x;
    if (idx >= N * 32) return;
    long long n = idx >> 5;
    int c = (int)(idx & 31);
    float dn = fmaxf(den[n], 1e-16f);
    out[idx] = out[idx] / dn + b2[c];
}

// ---------------------------------------------------------------- launcher

extern "C" void kernel_launch(void* const* d_in, const int* in_sizes, int n_in,
                              void* d_out, int out_size, void* d_ws, size_t ws_size,
                              hipStream_t stream) {
    const float* x      = (const float*)d_in[0];
    const float* W1     = (const float*)d_in[1];
    const float* a_src1 = (const float*)d_in[2];
    const float* a_dst1 = (const float*)d_in[3];
    const float* b1     = (const float*)d_in[4];
    const float* W2     = (const float*)d_in[5];
    const float* a_src2 = (const float*)d_in[6];
    const float* a_dst2 = (const float*)d_in[7];
    const float* b2     = (const float*)d_in[8];
    const long long* ei = (const long long*)d_in[9];   // reference dtype: int64

    const long long N = in_sizes[0] / 128;
    const long long E = in_sizes[9] / 2;
    const long long Et = E + N;                        // edges incl. self-loops

    // workspace layout (floats)
    float* w     = (float*)d_ws;
    float* h1    = w;                  // N*128  pre-activation layer-1 features
    float* acc1  = h1 + N * 128;       // N*128  numerator accum -> becomes ELU output
    float* h2    = acc1 + N * 128;     // N*32   layer-2 features
    float* als1  = h2 + N * 32;        // N*4
    float* ald1  = als1 + N * 4;       // N*4
    float* m1    = ald1 + N * 4;       // N*4
    float* den1  = m1 + N * 4;         // N*4
    float* als2  = den1 + N * 4;       // N
    float* ald2  = als2 + N;           // N
    float* m2    = ald2 + N;           // N
    float* den2  = m2 + N;             // N
    float* pbuf1 = den2 + N;           // Et*4  softmax numerators layer 1
    float* pbuf2 = pbuf1 + Et * 4;     // Et    softmax numerators layer 2
    float* outf  = (float*)d_out;      // N*32

    const int T = 256;
    auto gsz = [](long long n, int t) { return (unsigned)((n + t - 1) / t); };

    // init accumulators each call (graph replays don't re-zero)
    fill_f32<<<gsz(N * 128, T), T, 0, stream>>>(acc1, 0.0f, N * 128);
    fill_f32<<<gsz(N * 4, T),   T, 0, stream>>>(den1, 0.0f, N * 4);
    fill_f32<<<gsz(N * 4, T),   T, 0, stream>>>(m1, -1e30f, N * 4);
    fill_f32<<<gsz(N * 32, T),  T, 0, stream>>>(outf, 0.0f, N * 32);
    fill_f32<<<gsz(N, T),       T, 0, stream>>>(den2, 0.0f, N);
    fill_f32<<<gsz(N, T),       T, 0, stream>>>(m2, -1e30f, N);

    const unsigned rowTiles = (unsigned)((N + 15) / 16);

    // ---- layer 1 ----
    gemm1_wmma<<<rowTiles, 256, 0, stream>>>(x, W1, h1, N);
    logits1<<<gsz(N, T), T, 0, stream>>>(h1, a_src1, a_dst1, als1, ald1, N);
    edge_max1<<<gsz(Et * 4, T), T, 0, stream>>>(ei, als1, ald1, m1, E, N);
    edge_p1<<<gsz(Et * 4, T), T, 0, stream>>>(ei, als1, ald1, m1, pbuf1, den1, E, N);
    edge_scatter1<<<gsz(Et * 128, T), T, 0, stream>>>(ei, pbuf1, h1, acc1, E, N);
    norm1<<<gsz(N * 128, T), T, 0, stream>>>(acc1, den1, b1, N);   // acc1 now = ELU(layer-1 out)

    // ---- layer 2 ----
    gemm2_wmma<<<rowTiles, 64, 0, stream>>>(acc1, W2, h2, N);
    logits2<<<gsz(N, T), T, 0, stream>>>(h2, a_src2, a_dst2, als2, ald2, N);
    edge_max2<<<gsz(Et, T), T, 0, stream>>>(ei, als2, ald2, m2, E, N);
    edge_p2<<<gsz(Et, T), T, 0, stream>>>(ei, als2, ald2, m2, pbuf2, den2, E, N);
    edge_scatter2<<<gsz(Et * 32, T), T, 0, stream>>>(ei, pbuf2, h2, outf, E, N);
    norm2<<<gsz(N * 32, T), T, 0, stream>>>(outf, den2, b2, N);
}